// CrossRangePenaltyLoss_77988016161298
// MI455X (gfx1250) — compile-verified
//
#include <hip/hip_runtime.h>
#include <cstdint>
#include <cstddef>

#define NUM_CLASSES 388
#define STAGED 384                         // floats per row staged via async DMA (3 x 512B)
#define TAIL (NUM_CLASSES - STAGED)        // 4 trailing floats read directly
#define WAVES_PER_BLOCK 8
#define BLOCK_THREADS (WAVES_PER_BLOCK * 32)
#define GRID_BLOCKS 1024

__device__ __forceinline__ int range_id(int c) {
    // note_on 0-127, note_off 128-255, time_shift 256-355, velocity 356-387
    return (c <= 127) ? 0 : (c <= 255) ? 1 : (c <= 355) ? 2 : 3;
}

// Asynchronously copy 1536 bytes (384 floats) of one row into LDS.
// Each of the 32 lanes moves three 16B chunks at +0 / +512 / +1024.
// The instruction offset is added to BOTH the global and LDS addresses,
// so one (lds, global) base pair serves all three issues. ASYNCcnt += 3.
// th:TH_LOAD_NT — single-pass stream, don't let 407MB churn the 192MB L2.
__device__ __forceinline__ void async_stage_row(uint32_t lds_byte, const float* row, int lane) {
    uint64_t g = (uint64_t)(uintptr_t)row + (uint32_t)(lane << 4);
    uint32_t l = lds_byte + (uint32_t)(lane << 4);
    asm volatile(
        "global_load_async_to_lds_b128 %0, %1, off th:TH_LOAD_NT\n\t"
        "global_load_async_to_lds_b128 %0, %1, off offset:512 th:TH_LOAD_NT\n\t"
        "global_load_async_to_lds_b128 %0, %1, off offset:1024 th:TH_LOAD_NT"
        :: "v"(l), "v"(g) : "memory");
}

__device__ __forceinline__ void wait_async_le3() { asm volatile("s_wait_asynccnt 0x3" ::: "memory"); }
__device__ __forceinline__ void wait_async_le0() { asm volatile("s_wait_asynccnt 0x0" ::: "memory"); }

__global__ void __launch_bounds__(BLOCK_THREADS)
ce_range_main(const float* __restrict__ preds, const int* __restrict__ tgts,
              float* __restrict__ acc, int n_rows) {
    __shared__ float stage[WAVES_PER_BLOCK][2][STAGED];   // 24576 B, per-wave double buffer
    __shared__ float red_loss[WAVES_PER_BLOCK];
    __shared__ float red_mis[WAVES_PER_BLOCK];

    const int tid  = threadIdx.x;
    const int w    = tid >> 5;
    const int lane = tid & 31;
    const int gw   = blockIdx.x * WAVES_PER_BLOCK + w;    // one wave == one row stream
    const int nw   = gridDim.x * WAVES_PER_BLOCK;

    // Low 32 bits of a generic pointer into LDS == byte offset in wave's LDS space.
    const uint32_t lds0 = (uint32_t)(uintptr_t)&stage[w][0][0];
    const uint32_t lds1 = (uint32_t)(uintptr_t)&stage[w][1][0];

    float acc_loss = 0.0f;
    float acc_mis  = 0.0f;

    if (gw < n_rows) {                                    // wave-uniform branch (EXEC stays full)
        async_stage_row(lds0, preds + (size_t)gw * NUM_CLASSES, lane);
        int i = 0;
        for (int r = gw; r < n_rows; r += nw, ++i) {
            const int rn = r + nw;
            const bool more = (rn < n_rows);              // wave-uniform
            const uint32_t lds_nxt = (i & 1) ? lds0 : lds1;
            if (more) {
                async_stage_row(lds_nxt, preds + (size_t)rn * NUM_CLASSES, lane);
                wait_async_le3();                         // oldest 3 (current stage) complete
            } else {
                wait_async_le0();
            }
            const float* buf = (i & 1) ? &stage[w][1][0] : &stage[w][0][0];

            // Bank-conflict-free strided reads: lane l takes elements l, l+32, ..., l+352.
            float v[12];
            #pragma unroll
            for (int k = 0; k < 12; ++k) v[k] = buf[lane + (k << 5)];
            const bool has_tail = lane < TAIL;            // elements 384..387 straight from global
            float tv = 0.0f;
            if (has_tail)
                tv = __builtin_nontemporal_load(&preds[(size_t)r * NUM_CLASSES + STAGED + lane]);

            // Lane-local max / argmax (first-occurrence tie-break, ascending index order)
            float bm = v[0];
            int   bi = lane;
            #pragma unroll
            for (int k = 1; k < 12; ++k) {
                const int idx = lane + (k << 5);
                if (v[k] > bm) { bm = v[k]; bi = idx; }
            }
            if (has_tail && tv > bm) { bm = tv; bi = STAGED + lane; }

            // wave32 xor-shuffle reduction for (max, argmax)
            #pragma unroll
            for (int off = 16; off > 0; off >>= 1) {
                const float om = __shfl_xor(bm, off, 32);
                const int   oi = __shfl_xor(bi, off, 32);
                if (om > bm || (om == bm && oi < bi)) { bm = om; bi = oi; }
            }

            // sum of exp(x - max)
            float s = 0.0f;
            #pragma unroll
            for (int k = 0; k < 12; ++k) s += __expf(v[k] - bm);
            if (has_tail) s += __expf(tv - bm);
            #pragma unroll
            for (int off = 16; off > 0; off >>= 1) s += __shfl_xor(s, off, 32);

            if (lane == 0) {
                const int   t  = __builtin_nontemporal_load(&tgts[r]);
                const float xt = (t < STAGED)
                                   ? buf[t]
                                   : __builtin_nontemporal_load(&preds[(size_t)r * NUM_CLASSES + t]);
                acc_loss += bm + __logf(s) - xt;          // -log_softmax at target
                acc_mis  += (range_id(bi) != range_id(t)) ? 1.0f : 0.0f;
            }
        }
    }

    if (lane == 0) { red_loss[w] = acc_loss; red_mis[w] = acc_mis; }
    __syncthreads();
    if (tid == 0) {
        float sl = 0.0f, sm = 0.0f;
        #pragma unroll
        for (int k = 0; k < WAVES_PER_BLOCK; ++k) { sl += red_loss[k]; sm += red_mis[k]; }
        atomicAdd(&acc[0], sl);                           // 2 atomics per block total
        atomicAdd(&acc[1], sm);
    }
}

__global__ void zero_acc(float* acc) {
    acc[0] = 0.0f;
    acc[1] = 0.0f;
}

__global__ void finalize_out(const float* __restrict__ acc, float* __restrict__ out, float inv_n) {
    out[0] = acc[0] * inv_n + 2.0f * acc[1] * inv_n;      // mean CE + PENALTY_FACTOR * mean mismatch
}

extern "C" void kernel_launch(void* const* d_in, const int* in_sizes, int n_in,
                              void* d_out, int out_size, void* d_ws, size_t ws_size,
                              hipStream_t stream) {
    const float* preds = (const float*)d_in[0];
    const int*   tgts  = (const int*)d_in[1];
    float* out = (float*)d_out;
    float* acc = (float*)d_ws;                            // 2 floats of scratch
    const int n_rows = in_sizes[1];                       // targets count == row count

    zero_acc<<<1, 1, 0, stream>>>(acc);
    ce_range_main<<<GRID_BLOCKS, BLOCK_THREADS, 0, stream>>>(preds, tgts, acc, n_rows);
    finalize_out<<<1, 1, 0, stream>>>(acc, out, 1.0f / (float)n_rows);
}